// Extract_RPN_Proposals_4827543240777
// MI455X (gfx1250) — compile-verified
//
#include <hip/hip_runtime.h>
#include <hip/hip_bf16.h>

// ---------------------------------------------------------------------------
// Extract_RPN_Proposals for MI455X (gfx1250, wave32).
//
// Phase A (decode_assign_compact): decode boxes, WMMA-f32 (16x16x4) tile GT
//   assignment with division-free cross-multiplied argmax, ballot-compact
//   candidates (conf > 0.5) into workspace.
// Phase B (nms_persistent): one 1024-thread workgroup runs the 300 serial NMS
//   rounds; lazy suppression fused into the argmax scan; division-free IoU
//   threshold test; wave32 butterfly argmax (2 barriers/round).
// ---------------------------------------------------------------------------

#define GRID_SZ     192
#define A_PER       9
#define NTOT        (GRID_SZ * GRID_SZ * A_PER)   // 331776, divisible by 256
#define NGT         64
#define NMS_MAXSEL  300
#define CONF_THRES  0.5f
#define NMS_IOU_T   0.7f
#define BG_IOU_T    0.5f
#define NUM_CLS_F   20.0f
#define NEGV        (-1000000000.0f)
#define NEG_HALF    (-500000000.0f)

static_assert(NTOT % 256 == 0, "block tiling assumption");

typedef __attribute__((ext_vector_type(2))) float v2f;
typedef __attribute__((ext_vector_type(8))) float v8f;

// ---- CDNA5 async global->LDS copy (ASYNCcnt-tracked) ----------------------
__device__ __forceinline__ void async_g2lds_b128(void* lds, const void* g) {
  // Low 32 bits of a generic pointer to LDS carry the LDS byte address.
  unsigned loff = (unsigned)(unsigned long long)lds;
  asm volatile("global_load_async_to_lds_b128 %0, %1, off"
               :: "v"(loff), "v"(g) : "memory");
}
__device__ __forceinline__ void wait_async0() {
  asm volatile("s_wait_asynccnt 0" ::: "memory");
}

// ---------------------------------------------------------------------------
// Kernel 0: zero the compaction counter (ws is not re-poisoned between runs).
// ---------------------------------------------------------------------------
__global__ void init_counter_kernel(int* __restrict__ counter) {
  if (threadIdx.x == 0 && blockIdx.x == 0) *counter = 0;
}

// ---------------------------------------------------------------------------
// Kernel A: decode + GT assignment (WMMA f32 16x16x4 area-sum tiles) + compact
// ---------------------------------------------------------------------------
__global__ __launch_bounds__(256) void decode_assign_compact(
    const float* __restrict__ confs,   const float* __restrict__ deltas,
    const float* __restrict__ anchors, const float* __restrict__ gt_objs,
    const float* __restrict__ gt_cls,
    float* __restrict__ cand_score, float* __restrict__ cand_box,
    float* __restrict__ cand_cls,   int* __restrict__ counter)
{
  __shared__ float s_gtraw[NGT * 4];   // staged via async copy: (cx,cy,w,h)
  __shared__ float s_gtclsr[NGT];      // staged via async copy
  __shared__ float s_gbox[NGT * 4];    // GT xymm
  __shared__ float s_garea[NGT];
  __shared__ float s_abox[256 * 4];    // anchor xymm
  __shared__ float s_aarea[256];
  __shared__ float s_assigned[256];

  const int tid = threadIdx.x;
  const int gid = blockIdx.x * 256 + tid;
  const int lane = tid & 31;
  const int wv   = tid >> 5;
  const int half = lane >> 4;          // 0: lanes 0-15, 1: lanes 16-31
  const int n16  = lane & 15;

  // ---- stage GT data into LDS asynchronously (80 lanes x 16B) ----
  if (tid < NGT) {
    async_g2lds_b128(&s_gtraw[tid * 4], gt_objs + tid * 4);
  } else if (tid < NGT + 16) {
    async_g2lds_b128(&s_gtclsr[(tid - NGT) * 4], gt_cls + (tid - NGT) * 4);
  }

  // ---- decode this thread's anchor (overlaps with async copy) ----
  const float conf = confs[gid];
  const float4 d  = ((const float4*)deltas)[gid];
  const float4 an = ((const float4*)anchors)[gid];
  const float cx = d.x * an.z + an.x;
  const float cy = d.y * an.w + an.y;
  const float w  = expf(d.z) * an.z;
  const float h  = expf(d.w) * an.w;
  const float x0 = cx - 0.5f * w, y0 = cy - 0.5f * h;
  const float x1 = cx + 0.5f * w, y1 = cy + 0.5f * h;
  s_abox[tid * 4 + 0] = x0; s_abox[tid * 4 + 1] = y0;
  s_abox[tid * 4 + 2] = x1; s_abox[tid * 4 + 3] = y1;
  s_aarea[tid] = (x1 - x0) * (y1 - y0);

  wait_async0();
  __syncthreads();

  // ---- convert GT to xymm + area ----
  if (tid < NGT) {
    const float gx = s_gtraw[tid * 4 + 0], gy = s_gtraw[tid * 4 + 1];
    const float gw = s_gtraw[tid * 4 + 2], gh = s_gtraw[tid * 4 + 3];
    const float gx0 = gx - 0.5f * gw, gy0 = gy - 0.5f * gh;
    const float gx1 = gx + 0.5f * gw, gy1 = gy + 0.5f * gh;
    s_gbox[tid * 4 + 0] = gx0; s_gbox[tid * 4 + 1] = gy0;
    s_gbox[tid * 4 + 2] = gx1; s_gbox[tid * 4 + 3] = gy1;
    s_garea[tid] = (gx1 - gx0) * (gy1 - gy0);
  }
  __syncthreads();

  // ---- GT assignment: per wave, two 16-anchor tiles vs 4 x 16-GT tiles ----
  // area_a[m] + area_b[n] for a 16x16 tile == [area_a,1] x [1;area_b] : one
  // V_WMMA_F32_16X16X4_F32 per tile. C layout: lane n = lane&15, row m = r +
  // 8*(lane>>4) for VGPR r. Control flow is uniform here (EXEC all ones).
  //
  // Argmax over GT boxes is done division-free: keep (num, den) of the best
  // IoU fraction; n1/d1 > n2/d2  <=>  n1*d2 > n2*d1 since all dens > 0.
  for (int t = 0; t < 2; ++t) {
    const int abase = wv * 32 + t * 16;  // 8 waves * 2 * 16 = 256 anchors
    // A frag (16x4 f32): lanes 0-15 hold K=0..1 -> (area_a, 1); lanes 16-31
    // hold K=2..3 -> (0, 0).
    const float a0 = (half == 0) ? s_aarea[abase + n16] : 0.0f;
    const float a1 = (half == 0) ? 1.0f : 0.0f;

    float bnum[8], bden[8];
    int   bn[8];
#pragma unroll
    for (int r = 0; r < 8; ++r) { bnum[r] = -1.0f; bden[r] = 1.0f; bn[r] = 0; }

#pragma unroll
    for (int g = 0; g < 4; ++g) {
      // B frag (4x16 f32): lanes 0-15 rows K=0..1 -> (1, area_b[n]); lanes
      // 16-31 rows K=2..3 -> (0, 0).
      const float b0 = (half == 0) ? 1.0f : 0.0f;
      const float b1 = (half == 0) ? s_garea[g * 16 + n16] : 0.0f;
      v8f S;
#if __has_builtin(__builtin_amdgcn_wmma_f32_16x16x4_f32)
      {
        v2f Af = {a0, a1};
        v2f Bf = {b0, b1};
        v8f C  = {0.f, 0.f, 0.f, 0.f, 0.f, 0.f, 0.f, 0.f};
        S = __builtin_amdgcn_wmma_f32_16x16x4_f32(
            /*neg_a=*/false, Af, /*neg_b=*/false, Bf,
            /*c_mod=*/(short)0, C, /*reuse_a=*/false, /*reuse_b=*/false);
      }
#else
      {
#pragma unroll
        for (int r = 0; r < 8; ++r)
          S[r] = s_aarea[abase + r + half * 8] + s_garea[g * 16 + n16];
      }
#endif
      const int n = g * 16 + n16;
      const float gx0 = s_gbox[n * 4 + 0], gy0 = s_gbox[n * 4 + 1];
      const float gx1 = s_gbox[n * 4 + 2], gy1 = s_gbox[n * 4 + 3];
#pragma unroll
      for (int r = 0; r < 8; ++r) {
        const int m = abase + r + half * 8;
        const float ax0 = s_abox[m * 4 + 0], ay0 = s_abox[m * 4 + 1];
        const float ax1 = s_abox[m * 4 + 2], ay1 = s_abox[m * 4 + 3];
        const float tlx = fmaxf(ax0, gx0), tly = fmaxf(ay0, gy0);
        const float brx = fminf(ax1, gx1), bry = fminf(ay1, gy1);
        const float iw = fmaxf(brx - tlx, 0.0f);
        const float ih = fmaxf(bry - tly, 0.0f);
        const float inter = iw * ih;
        const float den = S[r] - inter + 1e-9f;   // > 0 always
        // inter/den > bnum/bden  <=>  inter*bden > bnum*den
        if (inter * bden[r] > bnum[r] * den) {
          bnum[r] = inter; bden[r] = den; bn[r] = n;
        }
      }
    }

    // Max-reduce over the 16 lanes of each half (argmax over n, first-index
    // tie-break like jnp.argmax). shfl_xor with masks 8..1 stays in-half.
#pragma unroll
    for (int r = 0; r < 8; ++r) {
      float num = bnum[r], den = bden[r];
      int   n = bn[r];
#pragma unroll
      for (int mk = 8; mk >= 1; mk >>= 1) {
        const float onum = __shfl_xor(num, mk, 32);
        const float oden = __shfl_xor(den, mk, 32);
        const int   on   = __shfl_xor(n, mk, 32);
        const float lhs = onum * den;   // other vs mine, cross-multiplied
        const float rhs = num * oden;
        if (lhs > rhs || (lhs == rhs && on < n)) {
          num = onum; den = oden; n = on;
        }
      }
      if (n16 == 0) {
        // best_iou < 0.5  <=>  num < 0.5*den   (den > 0)
        s_assigned[abase + half * 8 + r] =
            (num < BG_IOU_T * den) ? NUM_CLS_F : s_gtclsr[n];
      }
    }
  }
  __syncthreads();

  // ---- wave32 ballot-aggregated candidate compaction ----
  const bool pass = conf > CONF_THRES;
  const unsigned mask = (unsigned)__ballot(pass);
  int base = 0;
  if (mask) {
    const int leader = __ffs(mask) - 1;
    int b = 0;
    if (lane == leader) b = atomicAdd(counter, __popc(mask));
    base = __shfl(b, leader, 32);
  }
  if (pass) {
    const int pos = base + __popc(mask & ((1u << lane) - 1u));
    cand_score[pos] = conf;                       // score == conf for survivors
    cand_box[pos * 4 + 0] = y0;                   // yxminmax order
    cand_box[pos * 4 + 1] = x0;
    cand_box[pos * 4 + 2] = y1;
    cand_box[pos * 4 + 3] = x1;
    cand_cls[pos] = s_assigned[tid];
  }
}

// ---------------------------------------------------------------------------
// Kernel B: persistent single-workgroup NMS, lazy suppression fused in argmax.
// ---------------------------------------------------------------------------
__global__ __launch_bounds__(1024) void nms_persistent(
    const float* __restrict__ cand_box, const float* __restrict__ cand_cls,
    float* __restrict__ cand_score, const int* __restrict__ counter,
    float* __restrict__ out)
{
  __shared__ float wval[32];
  __shared__ int   widx[32];
  __shared__ float sbox[4];
  __shared__ int   s_have;

  const int tid  = threadIdx.x;
  const int lane = tid & 31;
  const int wv   = tid >> 5;
  const int M = *counter;
  const float4* __restrict__ boxes4 = (const float4*)cand_box;

  if (tid == 0) s_have = 0;
  if (tid < 4)  sbox[tid] = 0.0f;
  __syncthreads();

  for (int t = 0; t < NMS_MAXSEL; ++t) {
    // Broadcast last round's winner (written before the trailing barrier).
    const bool  have = (s_have != 0);
    const float b0 = sbox[0], b1 = sbox[1], b2 = sbox[2], b3 = sbox[3];
    const float larea = (b2 - b0) * (b3 - b1);

    // Fused scan: apply suppression vs. last winner while finding local max.
    // Division-free: iou > 0.7  <=>  inter > 0.7*(la+ca-inter+eps).
    float best = -3.0e38f;
    int   bidx = -1;
    for (int j = tid; j < M; j += 1024) {
      if (j + 1024 < M) {
        __builtin_prefetch(&boxes4[j + 1024], 0, 0);   // global_prefetch_b8
      }
      float s = cand_score[j];
      if (s > NEG_HALF) {
        if (have) {
          const float4 cb = boxes4[j];
          const float tly = fmaxf(b0, cb.x), tlx = fmaxf(b1, cb.y);
          const float bry = fminf(b2, cb.z), brx = fminf(b3, cb.w);
          const float ih = fmaxf(bry - tly, 0.0f);
          const float iw = fmaxf(brx - tlx, 0.0f);
          const float inter = ih * iw;
          const float carea = (cb.z - cb.x) * (cb.w - cb.y);
          if (inter > NMS_IOU_T * (larea + carea - inter + 1e-9f)) {
            s = NEGV; cand_score[j] = NEGV;
          }
        }
        if (s > best) { best = s; bidx = j; }
      }
    }

    // Wave32 butterfly argmax (first-index tie-break), then 32 partials.
#pragma unroll
    for (int mk = 16; mk >= 1; mk >>= 1) {
      const float ov = __shfl_xor(best, mk, 32);
      const int   oi = __shfl_xor(bidx, mk, 32);
      if (oi != -1 &&
          (ov > best || (ov == best && (bidx == -1 || oi < bidx)))) {
        best = ov; bidx = oi;
      }
    }
    if (lane == 0) { wval[wv] = best; widx[wv] = bidx; }
    __syncthreads();

    // Wave 0 reduces the 32 per-wave partials and finalizes the round.
    if (tid < 32) {
      float v = wval[tid];
      int   i = widx[tid];
#pragma unroll
      for (int mk = 16; mk >= 1; mk >>= 1) {
        const float ov = __shfl_xor(v, mk, 32);
        const int   oi = __shfl_xor(i, mk, 32);
        if (oi != -1 && (ov > v || (ov == v && (i == -1 || oi < i)))) {
          v = ov; i = oi;
        }
      }
      if (tid == 0) {
        const bool valid = (i >= 0) && (v > NEG_HALF);
        out[t] = valid ? v : 0.0f;
        if (valid) {
          const float4 cb = boxes4[i];
          out[NMS_MAXSEL + 4 * t + 0] = cb.x;
          out[NMS_MAXSEL + 4 * t + 1] = cb.y;
          out[NMS_MAXSEL + 4 * t + 2] = cb.z;
          out[NMS_MAXSEL + 4 * t + 3] = cb.w;
          out[5 * NMS_MAXSEL + t] = cand_cls[i];
          sbox[0] = cb.x; sbox[1] = cb.y; sbox[2] = cb.z; sbox[3] = cb.w;
          s_have = 1;
          cand_score[i] = NEGV;     // scores.at[i].set(NEG)
        } else {
          out[NMS_MAXSEL + 4 * t + 0] = 0.0f;
          out[NMS_MAXSEL + 4 * t + 1] = 0.0f;
          out[NMS_MAXSEL + 4 * t + 2] = 0.0f;
          out[NMS_MAXSEL + 4 * t + 3] = 0.0f;
          out[5 * NMS_MAXSEL + t] = NUM_CLS_F;
          s_have = 0;
        }
      }
    }
    __syncthreads();
  }
}

// ---------------------------------------------------------------------------
extern "C" void kernel_launch(void* const* d_in, const int* in_sizes, int n_in,
                              void* d_out, int out_size, void* d_ws,
                              size_t ws_size, hipStream_t stream) {
  const float* confs   = (const float*)d_in[0];
  const float* deltas  = (const float*)d_in[1];
  const float* anchors = (const float*)d_in[2];
  const float* gt_objs = (const float*)d_in[3];
  const float* gt_cls  = (const float*)d_in[4];
  float* out = (float*)d_out;

  // ws layout: [counter pad 256B][score N][box 4N][cls N]  (~8 MB)
  int*   counter    = (int*)d_ws;
  float* cand_score = (float*)((char*)d_ws + 256);
  float* cand_box   = cand_score + NTOT;
  float* cand_cls   = cand_box + 4 * NTOT;

  init_counter_kernel<<<1, 32, 0, stream>>>(counter);
  decode_assign_compact<<<NTOT / 256, 256, 0, stream>>>(
      confs, deltas, anchors, gt_objs, gt_cls,
      cand_score, cand_box, cand_cls, counter);
  nms_persistent<<<1, 1024, 0, stream>>>(cand_box, cand_cls, cand_score,
                                         counter, out);
}